// SphericalConvCodec_74053826118353
// MI455X (gfx1250) — compile-verified
//
#include <hip/hip_runtime.h>
#include <hip/hip_bf16.h>

typedef __attribute__((ext_vector_type(2))) float v2f;
typedef __attribute__((ext_vector_type(8))) float v8f;

#define N_SAMPLES 2048
#define NPOINTS   (N_SAMPLES * 16)   // 32768 latent points
#define CB_N      8192
#define LD        12

// ---------------------------------------------------------------------------
// Kernel 0: per-entry squared norm of the codebook (|c|^2 term of d^2)
// ---------------------------------------------------------------------------
__global__ void k_cnorm(const float* __restrict__ cb, float* __restrict__ cn2) {
  int e = blockIdx.x * blockDim.x + threadIdx.x;
  if (e < CB_N) {
    const float* c = cb + (size_t)e * LD;
    float s = 0.f;
#pragma unroll
    for (int i = 0; i < LD; ++i) s += c[i] * c[i];
    cn2[e] = s;
  }
}

// ---------------------------------------------------------------------------
// Kernel 1: encoder (3 strided convs + L2 normalize). One block per sample.
// Writes normalized continuous to d_out and point-major latents to ws.
// ---------------------------------------------------------------------------
__global__ void k_encoder(const float* __restrict__ x,
                          const float* __restrict__ w1, const float* __restrict__ b1,
                          const float* __restrict__ w2, const float* __restrict__ b2,
                          const float* __restrict__ w3, const float* __restrict__ b3,
                          float* __restrict__ out_cont, float* __restrict__ lat) {
  __shared__ float sx[1024];
  __shared__ float sh1[16 * 256];
  __shared__ float sh2[32 * 64];
  __shared__ float sc[12 * 16];
  __shared__ float sscale[16];
  const int n = blockIdx.x, tid = threadIdx.x;

  const float* xr = x + (size_t)n * 1024;
  for (int i = tid; i < 1024; i += 256) sx[i] = xr[i];
  __syncthreads();

  // h1[o,l] = relu(b1[o] + sum_k x[4l+k] w1[o,k]); thread owns l = tid
  {
    float x0 = sx[4 * tid], x1 = sx[4 * tid + 1], x2 = sx[4 * tid + 2], x3 = sx[4 * tid + 3];
#pragma unroll
    for (int o = 0; o < 16; ++o) {
      float v = b1[o] + x0 * w1[o * 4] + x1 * w1[o * 4 + 1] + x2 * w1[o * 4 + 2] + x3 * w1[o * 4 + 3];
      sh1[o * 256 + tid] = fmaxf(v, 0.f);
    }
  }
  __syncthreads();

  // h2[o,l] (32 x 64)
  for (int t = tid; t < 32 * 64; t += 256) {
    int o = t >> 6, l = t & 63;
    float v = b2[o];
#pragma unroll
    for (int c = 0; c < 16; ++c) {
      const float* wp = w2 + o * 64 + c * 4;
      const float* hp = sh1 + c * 256 + 4 * l;
      v += hp[0] * wp[0] + hp[1] * wp[1] + hp[2] * wp[2] + hp[3] * wp[3];
    }
    sh2[o * 64 + l] = fmaxf(v, 0.f);
  }
  __syncthreads();

  // continuous[c,l] (12 x 16)
  if (tid < 192) {
    int c = tid >> 4, l = tid & 15;
    float v = b3[c];
#pragma unroll
    for (int cc = 0; cc < 32; ++cc) {
      const float* wp = w3 + c * 128 + cc * 4;
      const float* hp = sh2 + cc * 64 + 4 * l;
      v += hp[0] * wp[0] + hp[1] * wp[1] + hp[2] * wp[2] + hp[3] * wp[3];
    }
    sc[c * 16 + l] = v;
  }
  __syncthreads();

  if (tid < 16) {
    float s = 0.f;
#pragma unroll
    for (int c = 0; c < 12; ++c) { float v = sc[c * 16 + tid]; s += v * v; }
    sscale[tid] = 1.f / fmaxf(sqrtf(s), 1e-12f);
  }
  __syncthreads();

  if (tid < 192) {
    int c = tid >> 4, l = tid & 15;
    float v = sc[c * 16 + l] * sscale[l];
    out_cont[(size_t)n * 192 + tid] = v;                 // (N,12,16) layout
    lat[((size_t)n * 16 + l) * 12 + c] = v;              // point-major for WMMA A
  }
}

// ---------------------------------------------------------------------------
// Kernel 2: VQ nearest-neighbor via V_WMMA_F32_16X16X4_F32.
// One wave = one 16-point tile; block stages 1024 codebook rows in LDS.
// score = |c|^2 - 2 * <x, c>  (matches the reference d2)
// ---------------------------------------------------------------------------
__global__ void __launch_bounds__(256) k_vq(const float* __restrict__ lat,
                                            const float* __restrict__ cbg,
                                            const float* __restrict__ cn2g,
                                            int* __restrict__ qidx) {
  __shared__ alignas(16) float scb[1024 * 12];   // 48 KB codebook stage
  __shared__ float scn[1024];                    // staged |c|^2
  const int tid  = threadIdx.x;
  const int lane = tid & 31;
  const int tile = blockIdx.x * 8 + (tid >> 5);  // 0..2047
  const int rowA = tile * 16 + (lane & 15);
  const int kHi  = (lane >> 4) << 1;             // lanes 0-15 -> K {k0,k0+1}; 16-31 -> {k0+2,k0+3}

  // A fragments for K-chunks {0..3},{4..7},{8..11}
  v2f a0 = *(const v2f*)(lat + (size_t)rowA * 12 + 0 + kHi);
  v2f a1 = *(const v2f*)(lat + (size_t)rowA * 12 + 4 + kHi);
  v2f a2 = *(const v2f*)(lat + (size_t)rowA * 12 + 8 + kHi);

  float mins[8]; int midx[8];
#pragma unroll
  for (int s = 0; s < 8; ++s) { mins[s] = 3.0e38f; midx[s] = 0; }

  for (int E0 = 0; E0 < CB_N; E0 += 1024) {
    const float4* src = (const float4*)(cbg + (size_t)E0 * 12);
    float4* dst = (float4*)scb;
    for (int i = tid; i < (1024 * 12) / 4; i += 256) dst[i] = src[i];
    for (int i = tid; i < 1024; i += 256) scn[i] = cn2g[E0 + i];
    __syncthreads();

    for (int e0 = 0; e0 < 1024; e0 += 16) {
      const float* bp = scb + (size_t)(e0 + (lane & 15)) * 12 + kHi;
      v2f bb0 = *(const v2f*)(bp + 0);
      v2f bb1 = *(const v2f*)(bp + 4);
      v2f bb2 = *(const v2f*)(bp + 8);
      v8f d = {0.f, 0.f, 0.f, 0.f, 0.f, 0.f, 0.f, 0.f};
      d = __builtin_amdgcn_wmma_f32_16x16x4_f32(false, a0, false, bb0, (short)0, d, false, false);
      d = __builtin_amdgcn_wmma_f32_16x16x4_f32(false, a1, false, bb1, (short)0, d, false, false);
      d = __builtin_amdgcn_wmma_f32_16x16x4_f32(false, a2, false, bb2, (short)0, d, false, false);
      float cn = scn[e0 + (lane & 15)];
      int   eg = E0 + e0 + (lane & 15);
#pragma unroll
      for (int s = 0; s < 8; ++s) {               // VGPR s: rows s (lanes 0-15), s+8 (lanes 16-31)
        float sc = cn - 2.0f * d[s];
        bool  t  = sc < mins[s];                  // strict < keeps earliest index in-lane
        mins[s] = t ? sc : mins[s];
        midx[s] = t ? eg : midx[s];
      }
    }
    __syncthreads();
  }

  // reduce across the 16 column-lanes of each half-wave; tie-break to lower idx
#pragma unroll
  for (int m = 1; m < 16; m <<= 1) {
#pragma unroll
    for (int s = 0; s < 8; ++s) {
      float os = __shfl_xor(mins[s], m, 32);
      int   oi = __shfl_xor(midx[s], m, 32);
      bool  t  = (os < mins[s]) || (os == mins[s] && oi < midx[s]);
      mins[s] = t ? os : mins[s];
      midx[s] = t ? oi : midx[s];
    }
  }
  if ((lane & 15) == 0) {
    int rowBase = tile * 16 + ((lane >> 4) << 3);  // lane 0 -> rows 0-7, lane 16 -> rows 8-15
#pragma unroll
    for (int s = 0; s < 8; ++s) qidx[rowBase + s] = midx[s];
  }
}

// ---------------------------------------------------------------------------
// Kernel 3: gather quantized vectors into (N,12,16) output
// ---------------------------------------------------------------------------
__global__ void k_gather(const int* __restrict__ qidx, const float* __restrict__ cb,
                         float* __restrict__ out_q) {
  int p = blockIdx.x * blockDim.x + threadIdx.x;
  if (p < NPOINTS) {
    int q = qidx[p];
    int n = p >> 4, l = p & 15;
    const float* c = cb + (size_t)q * 12;
    float* o = out_q + (size_t)n * 192 + l;
#pragma unroll
    for (int cc = 0; cc < 12; ++cc) o[cc * 16] = c[cc];
  }
}

// ---------------------------------------------------------------------------
// Kernel 4: decoder (3 transposed convs). One block per sample.
// ---------------------------------------------------------------------------
__global__ void k_decoder(const int* __restrict__ qidx, const float* __restrict__ cb,
                          const float* __restrict__ d1w, const float* __restrict__ d1b,
                          const float* __restrict__ d2w, const float* __restrict__ d2b,
                          const float* __restrict__ d3w, const float* __restrict__ d3b,
                          float* __restrict__ out) {
  __shared__ float sq[12 * 16];
  __shared__ float sg1[32 * 64];
  __shared__ float sg2[16 * 256];
  const int n = blockIdx.x, tid = threadIdx.x;

  if (tid < 192) {
    int c = tid >> 4, l = tid & 15;
    sq[tid] = cb[(size_t)qidx[n * 16 + l] * 12 + c];
  }
  __syncthreads();

  // g1[o, 4l+k] = relu(d1b[o] + sum_c q[c,l] d1w[c,o,k])   (32 x 64)
  for (int t = tid; t < 32 * 64; t += 256) {
    int o = t >> 6, j = t & 63;
    int l = j >> 2, k = j & 3;
    float v = d1b[o];
#pragma unroll
    for (int c = 0; c < 12; ++c) v += sq[c * 16 + l] * d1w[c * 128 + o * 4 + k];
    sg1[o * 64 + j] = fmaxf(v, 0.f);
  }
  __syncthreads();

  // g2[o, 4j+k]  (16 x 256)
  for (int t = tid; t < 16 * 256; t += 256) {
    int o = t >> 8, j2 = t & 255;
    int j = j2 >> 2, k = j2 & 3;
    float v = d2b[o];
#pragma unroll
    for (int c = 0; c < 32; ++c) v += sg1[c * 64 + j] * d2w[c * 64 + o * 4 + k];
    sg2[o * 256 + j2] = fmaxf(v, 0.f);
  }
  __syncthreads();

  // recon[4j2+k]  (1024)
  float* orow = out + (size_t)n * 1024;
  for (int j3 = tid; j3 < 1024; j3 += 256) {
    int j2 = j3 >> 2, k = j3 & 3;
    float v = d3b[0];
#pragma unroll
    for (int c = 0; c < 16; ++c) v += sg2[c * 256 + j2] * d3w[c * 4 + k];
    orow[j3] = v;
  }
}

// ---------------------------------------------------------------------------
extern "C" void kernel_launch(void* const* d_in, const int* in_sizes, int n_in,
                              void* d_out, int out_size, void* d_ws, size_t ws_size,
                              hipStream_t stream) {
  const float* x   = (const float*)d_in[0];
  const float* w1  = (const float*)d_in[1];
  const float* b1  = (const float*)d_in[2];
  const float* w2  = (const float*)d_in[3];
  const float* b2  = (const float*)d_in[4];
  const float* w3  = (const float*)d_in[5];
  const float* b3  = (const float*)d_in[6];
  const float* d1w = (const float*)d_in[7];
  const float* d1b = (const float*)d_in[8];
  const float* d2w = (const float*)d_in[9];
  const float* d2b = (const float*)d_in[10];
  const float* d3w = (const float*)d_in[11];
  const float* d3b = (const float*)d_in[12];
  const float* cb  = (const float*)d_in[13];

  float* out       = (float*)d_out;
  float* out_recon = out;                                // 2048*1024
  float* out_cont  = out + (size_t)N_SAMPLES * 1024;     // 2048*12*16
  float* out_quant = out_cont + (size_t)N_SAMPLES * 192; // 2048*12*16

  float* lat  = (float*)d_ws;                  // 32768*12 f32
  float* cn2  = lat + (size_t)NPOINTS * 12;    // 8192 f32
  int*   qidx = (int*)(cn2 + CB_N);            // 32768 i32

  k_cnorm  <<<CB_N / 256,          256, 0, stream>>>(cb, cn2);
  k_encoder<<<N_SAMPLES,           256, 0, stream>>>(x, w1, b1, w2, b2, w3, b3, out_cont, lat);
  k_vq     <<<NPOINTS / (16 * 8),  256, 0, stream>>>(lat, cb, cn2, qidx);
  k_gather <<<NPOINTS / 256,       256, 0, stream>>>(qidx, cb, out_quant);
  k_decoder<<<N_SAMPLES,           256, 0, stream>>>(qidx, cb, d1w, d1b, d2w, d2b, d3w, d3b, out_recon);
}